// _FeaturePropagationWithAttention_86165633892451
// MI455X (gfx1250) — compile-verified
//
#include <hip/hip_runtime.h>

typedef __attribute__((ext_vector_type(2))) float v2f;
typedef __attribute__((ext_vector_type(8))) float v8f;

#define B_  8
#define N_  16384
#define S_  2048
#define D1_ 128
#define D2_ 256
#define M_  (B_ * N_)   // 131072 total rows

// =====================================================================
// 1) Brute-force 3-NN + inverse-distance interpolation
//    grid: (N_/128, B_), block 128.  xyz2 staged in LDS; the s-loop index
//    is wave-uniform so LDS reads are broadcasts (no bank conflicts).
// =====================================================================
__global__ void knn_interp_kernel(const float* __restrict__ xyz1,
                                  const float* __restrict__ xyz2,
                                  const float* __restrict__ points2,
                                  float* __restrict__ interp) {
  __shared__ float xs[S_], ys[S_], zs[S_];
  __shared__ float w_sh[3][128];
  __shared__ int   i_sh[3][128];
  const int b = blockIdx.y;
  const int n = blockIdx.x * 128 + threadIdx.x;

  const float* x2 = xyz2 + (size_t)b * S_ * 3;
  for (int s = threadIdx.x; s < S_; s += 128) {
    xs[s] = x2[s * 3 + 0];
    ys[s] = x2[s * 3 + 1];
    zs[s] = x2[s * 3 + 2];
  }
  __syncthreads();

  const size_t qoff = ((size_t)b * N_ + n) * 3;
  const float qx = xyz1[qoff + 0];
  const float qy = xyz1[qoff + 1];
  const float qz = xyz1[qoff + 2];

  float d0 = 3.4e38f, d1 = 3.4e38f, d2 = 3.4e38f;
  int   i0 = 0, i1 = 0, i2 = 0;
  for (int s = 0; s < S_; ++s) {
    float dx = qx - xs[s], dy = qy - ys[s], dz = qz - zs[s];
    float d  = dx * dx + dy * dy + dz * dz;
    if (d < d0)      { d2 = d1; i2 = i1; d1 = d0; i1 = i0; d0 = d; i0 = s; }
    else if (d < d1) { d2 = d1; i2 = i1; d1 = d;  i1 = s; }
    else if (d < d2) { d2 = d;  i2 = s; }
  }
  const float r0 = 1.f / (d0 + 1e-8f);
  const float r1 = 1.f / (d1 + 1e-8f);
  const float r2 = 1.f / (d2 + 1e-8f);
  const float rs = r0 + r1 + r2;
  w_sh[0][threadIdx.x] = r0 / rs;
  w_sh[1][threadIdx.x] = r1 / rs;
  w_sh[2][threadIdx.x] = r2 / rs;
  i_sh[0][threadIdx.x] = i0;
  i_sh[1][threadIdx.x] = i1;
  i_sh[2][threadIdx.x] = i2;
  __syncthreads();

  // cooperative, channel-coalesced gather of the 3 neighbor feature rows
  const float* p2 = points2 + (size_t)b * S_ * D2_;
  for (int p = 0; p < 128; ++p) {
    const int   np = blockIdx.x * 128 + p;
    const float w0 = w_sh[0][p], w1 = w_sh[1][p], w2 = w_sh[2][p];
    const int   j0 = i_sh[0][p], j1 = i_sh[1][p], j2 = i_sh[2][p];
    float* dst = interp + ((size_t)b * N_ + np) * D2_;
    for (int c = threadIdx.x; c < D2_; c += 128) {
      dst[c] = w0 * p2[(size_t)j0 * D2_ + c]
             + w1 * p2[(size_t)j1 * D2_ + c]
             + w2 * p2[(size_t)j2 * D2_ + c];
    }
  }
}

// =====================================================================
// 2) FP32 WMMA GEMM:  C[m,o] = sum_k A(m,k) * Wt[o,k] + bias[o]
//    A(m,k) = (k < split) ? A0[m*ldA0+k] : A1[m*ldA1+k-split]
//    (fuses the concat(points1, interp) for layer 1)
//    block = 256 threads = 8 waves in 4x2; block tile 64x64;
//    each wave: one 16-row strip x two 16-col tiles via
//    V_WMMA_F32_16X16X4_F32, K staged 64-wide through LDS.
//
//    A tile Xs[row][k]: consecutive k adjacent -> single ds_load_b64
//    per A fragment. W tile stored K-PAIRED: Wp[k/2][col][2] so each
//    B fragment (rows k,k+1 of one column) is one aligned ds_load_b64.
//    Pad 65 makes staging stores bank-conflict-free:
//    bank = (2*kp + 2*o + (k&1)) mod 64 is a permutation over the
//    64 threads sharing an o-row.
// =====================================================================
__global__ void wmma_gemm_kernel(const float* __restrict__ A0, int ldA0,
                                 const float* __restrict__ A1, int ldA1, int split,
                                 const float* __restrict__ Wt,
                                 const float* __restrict__ bias,
                                 float* __restrict__ Cout, int ldC,
                                 int K) {
  __shared__ float Xs[64][68];      // A tile: 64 rows x 64 k (+pad)
  __shared__ float Wp[32][65][2];   // B tile: 32 k-pairs x 64 cols (+pad), paired K
  const int tid  = threadIdx.x;
  const int lane = tid & 31;
  const int wave = tid >> 5;
  const int wrow = wave & 3;        // 16-row strip within block tile
  const int wcol = wave >> 2;       // 32-col strip within block tile
  const int m0 = blockIdx.x * 64;
  const int o0 = blockIdx.y * 64;

  v8f c0 = {};
  v8f c1 = {};

  for (int kc = 0; kc < K; kc += 64) {
    { // stage A (coalesced along k)
      const int k = tid & 63;
      for (int r = tid >> 6; r < 64; r += 4) {
        const int kk = kc + k;
        const int m  = m0 + r;
        Xs[r][k] = (kk < split) ? A0[(size_t)m * ldA0 + kk]
                                : A1[(size_t)m * ldA1 + (kk - split)];
      }
    }
    { // stage W^T into paired-K layout (coalesced along k of the weight row)
      const int k = tid & 63;
      for (int o = tid >> 6; o < 64; o += 4) {
        Wp[k >> 1][o][k & 1] = Wt[(size_t)(o0 + o) * K + kc + k];
      }
    }
    // prefetch next K-chunk into GL2 while this chunk is consumed
    if (kc + 64 < K) {
      const int knext = kc + 64 + (tid & 63);
      const int rnext = tid >> 6;
      const float* pa = (knext < split)
                          ? &A0[(size_t)(m0 + rnext) * ldA0 + knext]
                          : &A1[(size_t)(m0 + rnext) * ldA1 + (knext - split)];
      __builtin_prefetch(pa, 0, 3);
      __builtin_prefetch(&Wt[(size_t)(o0 + rnext) * K + knext], 0, 3);
    }
    __syncthreads();

    // ISA 7.12.2 f32 layouts:
    //  A 16x4:  lanes 0-15 hold K={0,1}, lanes 16-31 hold K={2,3}
    //  B 4x16:  VGPR i -> row (i + 2*half), col = lane&15
    const int ml   = wrow * 16 + (lane & 15);
    const int ksel = (lane >> 4) << 1;           // 0 (lo half) or 2 (hi half)
    const int col0 = wcol * 32 + (lane & 15);
#pragma unroll
    for (int kk = 0; kk < 16; ++kk) {
      const int kb = kk * 4;
      const int kp = (kb + ksel) >> 1;           // even -> 8B-aligned pair
      const v2f a  = *(const v2f*)(&Xs[ml][kb + ksel]);
      const v2f b0 = *(const v2f*)(&Wp[kp][col0][0]);
      const v2f b1 = *(const v2f*)(&Wp[kp][col0 + 16][0]);
      c0 = __builtin_amdgcn_wmma_f32_16x16x4_f32(false, a, false, b0,
                                                 (short)0, c0, false, false);
      c1 = __builtin_amdgcn_wmma_f32_16x16x4_f32(false, a, false, b1,
                                                 (short)0, c1, false, false);
    }
    __syncthreads();
  }

  // C/D layout: lane<16 -> col=lane, VGPR i = row i; lane>=16 -> rows +8
  const int chi    = (lane >> 4) * 8;
  const int ccol   = lane & 15;
  const int mbase  = m0 + wrow * 16 + chi;
  const int obase  = o0 + wcol * 32 + ccol;
  const float bv0  = bias[obase];
  const float bv1  = bias[obase + 16];
#pragma unroll
  for (int i = 0; i < 8; ++i) {
    Cout[(size_t)(mbase + i) * ldC + obase]      = c0[i] + bv0;
    Cout[(size_t)(mbase + i) * ldC + obase + 16] = c1[i] + bv1;
  }
}

// =====================================================================
// 3) Per-channel sum / sum-of-squares for train-mode BatchNorm
// =====================================================================
template <int C>
__global__ void stats_kernel(const float* __restrict__ X,
                             float* __restrict__ sum,
                             float* __restrict__ sumsq) {
  const int ROWS = 512;
  const int t    = threadIdx.x;      // 256 threads
  const int ch   = t % C;
  const int sub  = t / C;
  const int nsub = 256 / C;
  const int r0   = blockIdx.x * ROWS;
  float s = 0.f, sq = 0.f;
  for (int r = r0 + sub; r < r0 + ROWS; r += nsub) {
    const float v = X[(size_t)r * C + ch];
    s  += v;
    sq += v * v;
  }
  atomicAdd(&sum[ch], s);
  atomicAdd(&sumsq[ch], sq);
}

// =====================================================================
// 4) Apply BatchNorm (population stats) + ReLU in place
// =====================================================================
template <int C>
__global__ void bn_relu_kernel(float* __restrict__ X,
                               const float* __restrict__ sum,
                               const float* __restrict__ sumsq,
                               const float* __restrict__ g,
                               const float* __restrict__ be) {
  const size_t i  = (size_t)blockIdx.x * blockDim.x + threadIdx.x;
  const int    ch = (int)(i % C);
  const float mu  = sum[ch] * (1.f / (float)M_);
  const float var = sumsq[ch] * (1.f / (float)M_) - mu * mu;
  const float inv = rsqrtf(var + 1e-5f);
  const float v   = g[ch] * (X[i] - mu) * inv + be[ch];
  X[i] = v > 0.f ? v : 0.f;
}

// =====================================================================
// 5) Per-batch avg + max pooling over N (values are >= 0 post-ReLU, so
//    unsigned-bit atomicMax is order-preserving)
// =====================================================================
__global__ void pool_kernel(const float* __restrict__ X,
                            float* __restrict__ psum,
                            unsigned int* __restrict__ pmax) {
  const int ROWS = 512;                  // divides N_, so one batch per block
  const int t   = threadIdx.x;           // 256 threads
  const int ch  = t & 127;
  const int sub = t >> 7;
  const int r0  = blockIdx.x * ROWS;
  const int b   = r0 / N_;
  float s = 0.f, mx = 0.f;
  for (int r = r0 + sub; r < r0 + ROWS; r += 2) {
    const float v = X[(size_t)r * 128 + ch];
    s  += v;
    mx  = fmaxf(mx, v);
  }
  atomicAdd(&psum[b * 128 + ch], s);
  atomicMax(&pmax[b * 128 + ch], __float_as_uint(mx));
}

// =====================================================================
// 6) SE attention: scale = sigmoid(fc(avg) + fc(mx)), fc = relu(.@fa1^T)@fa2^T
//    Tiny: one block, 128 threads.
// =====================================================================
__global__ void attention_kernel(const float* __restrict__ psum,
                                 const unsigned int* __restrict__ pmax,
                                 const float* __restrict__ fa1,  // [16,128]
                                 const float* __restrict__ fa2,  // [128,16]
                                 float* __restrict__ scale) {
  __shared__ float va[128], vm[128], ha[16], hm[16];
  const int t = threadIdx.x;   // 128
  for (int b = 0; b < B_; ++b) {
    va[t] = psum[b * 128 + t] * (1.f / (float)N_);
    vm[t] = __uint_as_float(pmax[b * 128 + t]);
    __syncthreads();
    if (t < 16) {
      float s = 0.f;
      for (int c = 0; c < 128; ++c) s += va[c] * fa1[t * 128 + c];
      ha[t] = s > 0.f ? s : 0.f;
    } else if (t < 32) {
      const int j = t - 16;
      float s = 0.f;
      for (int c = 0; c < 128; ++c) s += vm[c] * fa1[j * 128 + c];
      hm[j] = s > 0.f ? s : 0.f;
    }
    __syncthreads();
    float o = 0.f;
    for (int j = 0; j < 16; ++j) o += (ha[j] + hm[j]) * fa2[t * 16 + j];
    scale[b * 128 + t] = 1.f / (1.f + expf(-o));
    __syncthreads();
  }
}

// =====================================================================
// 7) Apply channel attention in place
// =====================================================================
__global__ void scale_kernel(float* __restrict__ X,
                             const float* __restrict__ scale) {
  const size_t i  = (size_t)blockIdx.x * 256 + threadIdx.x;
  const int    ch = (int)(i & 127);
  const size_t m  = i >> 7;
  const int    b  = (int)(m / N_);
  X[i] *= scale[b * 128 + ch];
}

// =====================================================================
extern "C" void kernel_launch(void* const* d_in, const int* in_sizes, int n_in,
                              void* d_out, int out_size, void* d_ws, size_t ws_size,
                              hipStream_t stream) {
  (void)in_sizes; (void)n_in; (void)out_size; (void)ws_size;
  const float* xyz1    = (const float*)d_in[0];
  const float* xyz2    = (const float*)d_in[1];
  const float* points1 = (const float*)d_in[2];
  const float* points2 = (const float*)d_in[3];
  const float* w1      = (const float*)d_in[4];
  const float* b1      = (const float*)d_in[5];
  const float* g1      = (const float*)d_in[6];
  const float* be1     = (const float*)d_in[7];
  const float* w2      = (const float*)d_in[8];
  const float* b2      = (const float*)d_in[9];
  const float* g2      = (const float*)d_in[10];
  const float* be2     = (const float*)d_in[11];
  const float* fa1     = (const float*)d_in[12];
  const float* fa2     = (const float*)d_in[13];

  // workspace layout (floats)
  float* ws = (float*)d_ws;
  float*        stats1_sum = ws;                 // 256
  float*        stats1_sq  = ws + 256;           // 256
  float*        stats2_sum = ws + 512;           // 128
  float*        stats2_sq  = ws + 640;           // 128
  float*        psum       = ws + 768;           // B*128 = 1024
  unsigned int* pmax       = (unsigned int*)(ws + 1792); // 1024
  float*        scale      = ws + 2816;          // 1024
  float*        interp     = ws + 4096;          // M_*256
  float*        h1         = interp + (size_t)M_ * D2_;  // M_*256
  float*        out        = (float*)d_out;      // M_*128 (used as h2 scratch)

  // zero all atomic-accumulated scratch (graph-capture safe)
  hipMemsetAsync(d_ws, 0, 4096 * sizeof(float), stream);

  // 1) 3-NN interpolation -> interp [M,256]
  dim3 gk(N_ / 128, B_);
  knn_interp_kernel<<<gk, 128, 0, stream>>>(xyz1, xyz2, points2, interp);

  // 2) layer 1: concat(points1[.,128], interp[.,256]) @ w1^T + b1 -> h1 [M,256]
  dim3 gg1(M_ / 64, 256 / 64);
  wmma_gemm_kernel<<<gg1, 256, 0, stream>>>(points1, 128, interp, 256, 128,
                                            w1, b1, h1, 256, 384);
  stats_kernel<256><<<M_ / 512, 256, 0, stream>>>(h1, stats1_sum, stats1_sq);
  bn_relu_kernel<256><<<M_, 256, 0, stream>>>(h1, stats1_sum, stats1_sq, g1, be1);

  // 3) layer 2: h1 @ w2^T + b2 -> out [M,128]
  dim3 gg2(M_ / 64, 128 / 64);
  wmma_gemm_kernel<<<gg2, 256, 0, stream>>>(h1, 256, nullptr, 0, 256,
                                            w2, b2, out, 128, 256);
  stats_kernel<128><<<M_ / 512, 256, 0, stream>>>(out, stats2_sum, stats2_sq);
  bn_relu_kernel<128><<<M_ / 2, 256, 0, stream>>>(out, stats2_sum, stats2_sq, g2, be2);

  // 4) SE attention
  pool_kernel<<<M_ / 512, 256, 0, stream>>>(out, psum, pmax);
  attention_kernel<<<1, 128, 0, stream>>>(psum, pmax, fa1, fa2, scale);
  scale_kernel<<<M_ / 2, 256, 0, stream>>>(out, scale);
}